// GCNConvLocal_31842887533161
// MI455X (gfx1250) — compile-verified
//
#include <hip/hip_runtime.h>
#include <hip/hip_bf16.h>

// GCN conv for MI455X (gfx1250, wave32).
// Inputs (setup_inputs order): d_in[0]=x fp32 [N,256], d_in[1]=edge_index int64 [N,16],
// d_in[2]=W fp32 [256,256]. Output fp32 [N,256]. Workspace holds h_scaled [N,256] fp32.

#define GCN_N      100000
#define GCN_DIN    256
#define GCN_DOUT   256
#define GCN_MAXDEG 16

typedef __attribute__((ext_vector_type(2))) float v2f;
typedef __attribute__((ext_vector_type(8))) float v8f;

// LDS x-tile stride in floats: 256 data + 4 pad -> keeps 16B alignment,
// and row stride % 64 banks == 4 so the 16-lane column reads hit 16 distinct banks.
#define XSTRIDE 260

// ---------------------------------------------------------------------------
// Kernel 1: h_ws[n, o] = (sum_k x[n,k] * W[o,k]) * rsqrt(deg[n])
// Block = 256 threads (8 waves). Block M-tile = 32 node rows (LDS-staged).
// Wave w: M-subtile = (w>>2)*16, N-range = (w&3)*64 .. +63 (four 16x16 WMMA tiles).
// K loop: 64 steps of V_WMMA_F32_16X16X4_F32.
//
// WMMA fp32 fragment layout (per CDNA5 ISA 7.12.2, wave32):
//   A 16x4 : lane L<16 -> M=L, v0=K0,v1=K1 ; lane L+16 -> M=L, v0=K2,v1=K3
//   B 4x16 : lane L<16 -> N=L, v0=K0,v1=K1 ; lane L+16 -> N=L, v0=K2,v1=K3
//   C 16x16: VGPR v -> M=v (lanes 0-15), M=v+8 (lanes 16-31); N = lane%16
// ---------------------------------------------------------------------------
__global__ __launch_bounds__(256)
void gcn_gemm_wmma(const float* __restrict__ x,
                   const long long* __restrict__ edge_index,
                   const float* __restrict__ W,
                   float* __restrict__ h)
{
    __shared__ float xs[32 * XSTRIDE];   // 32 rows of x, padded
    __shared__ float isd[32];            // inv_sqrt_deg per row

    const int tid     = threadIdx.x;
    const int block_m = blockIdx.x * 32;

    // Cooperative load: 32 rows x 256 floats = 2048 float4 chunks, 8 per thread.
    {
        const float4* xg = (const float4*)(x + (size_t)block_m * GCN_DIN);
        #pragma unroll
        for (int it = 0; it < 8; ++it) {
            int c    = tid + it * 256;        // 0..2047
            int row  = c >> 6;                // /64 float4 per row
            int col4 = c & 63;
            float4 v = xg[(size_t)row * 64 + col4];
            *(float4*)(xs + row * XSTRIDE + col4 * 4) = v;
        }
    }
    // Degree -> inv sqrt, one thread per row.
    if (tid < 32) {
        const long long* er = edge_index + (size_t)(block_m + tid) * GCN_MAXDEG;
        int c = 1;
        #pragma unroll
        for (int k = 0; k < GCN_MAXDEG; ++k) c += (er[k] >= 0) ? 1 : 0;
        isd[tid] = rsqrtf((float)c);
    }
    __syncthreads();

    const int wave  = tid >> 5;
    const int lane  = tid & 31;
    const int msub  = wave >> 2;          // 0 or 1 -> which 16-row subtile
    const int n0    = (wave & 3) * 64;    // column base for this wave
    const int ncol  = lane & 15;          // N (and M for A) within tile
    const int khalf = lane >> 4;          // 0: K{0,1}, 1: K{2,3}

    const float* xtile = xs + msub * 16 * XSTRIDE;

    v8f acc[4];
    #pragma unroll
    for (int t = 0; t < 4; ++t) acc[t] = (v8f){};

    // Per-lane W row pointers: lane covers column n0+t*16+ncol, K offset 2*khalf.
    const float* wbase = W + (size_t)(n0 + ncol) * GCN_DIN + 2 * khalf;

    #pragma unroll 4
    for (int k = 0; k < GCN_DIN; k += 4) {
        const int acol = k + 2 * khalf;
        v2f a = *(const v2f*)(xtile + ncol * XSTRIDE + acol);
        #pragma unroll
        for (int t = 0; t < 4; ++t) {
            v2f b = *(const v2f*)(wbase + (size_t)t * 16 * GCN_DIN + k);
            acc[t] = __builtin_amdgcn_wmma_f32_16x16x4_f32(
                false, a, false, b, (short)0, acc[t], false, false);
        }
    }

    // Scale by inv_sqrt_deg and store h tile.
    const float* isdp = isd + msub * 16 + 8 * khalf;   // rows for this lane's C VGPRs
    const size_t row0 = (size_t)(block_m + msub * 16 + 8 * khalf);
    #pragma unroll
    for (int t = 0; t < 4; ++t) {
        float* op = h + row0 * GCN_DOUT + (n0 + t * 16 + ncol);
        #pragma unroll
        for (int v = 0; v < 8; ++v) {
            op[(size_t)v * GCN_DOUT] = acc[t][v] * isdp[v];
        }
    }
}

// ---------------------------------------------------------------------------
// Kernel 2: out[i] = (h[i] + sum_{e in edges(i), e>=0} h[e]) * rsqrt(deg[i])
// One block per node; 256 threads = one feature each (coalesced row reads).
// h is L2-resident (102 MB < 192 MB L2), so the random row gather runs at L2 bw.
// ---------------------------------------------------------------------------
__global__ __launch_bounds__(256)
void gcn_gather(const long long* __restrict__ edge_index,
                const float* __restrict__ h,
                float* __restrict__ out)
{
    __shared__ long long eidx[GCN_MAXDEG];

    const int i   = blockIdx.x;
    const int tid = threadIdx.x;

    if (tid < GCN_MAXDEG)
        eidx[tid] = edge_index[(size_t)i * GCN_MAXDEG + tid];
    __syncthreads();

    int cnt = 1;
    #pragma unroll
    for (int k = 0; k < GCN_MAXDEG; ++k) cnt += (eidx[k] >= 0) ? 1 : 0;
    const float invs = rsqrtf((float)cnt);

    float acc = h[(size_t)i * GCN_DOUT + tid];
    #pragma unroll
    for (int k = 0; k < GCN_MAXDEG; ++k) {
        long long e = eidx[k];
        if (e >= 0) acc += h[(size_t)e * GCN_DOUT + tid];
    }
    out[(size_t)i * GCN_DOUT + tid] = acc * invs;
}

extern "C" void kernel_launch(void* const* d_in, const int* in_sizes, int n_in,
                              void* d_out, int out_size, void* d_ws, size_t ws_size,
                              hipStream_t stream) {
    const float*     x  = (const float*)d_in[0];
    const long long* ei = (const long long*)d_in[1];
    const float*     W  = (const float*)d_in[2];
    float* out = (float*)d_out;
    float* h   = (float*)d_ws;                    // N*256 fp32 = 102.4 MB scratch

    (void)in_sizes; (void)n_in; (void)out_size; (void)ws_size;

    gcn_gemm_wmma<<<GCN_N / 32, 256, 0, stream>>>(x, ei, W, h);
    gcn_gather<<<GCN_N, 256, 0, stream>>>(ei, h, out);
}